// GateLoopOperator_90598040141880
// MI455X (gfx1250) — compile-verified
//
#include <hip/hip_runtime.h>

typedef __attribute__((ext_vector_type(2))) float v2f;
typedef __attribute__((ext_vector_type(8))) float v8f;
typedef __attribute__((ext_vector_type(4))) unsigned int u32x4;
typedef __attribute__((ext_vector_type(8))) int i32x8;
typedef __attribute__((ext_vector_type(4))) int i32x4;

// D = A(16x4) * B(4x16) + C ; fp32 WMMA (CDNA5 only)
#define WMMA4(a, b, c) \
  __builtin_amdgcn_wmma_f32_16x16x4_f32(false, (a), false, (b), (short)0, (c), false, false)

// ---------------- problem constants ----------------
constexpr int BATCH = 4, PCH = 2, TT = 32;
constexpr int L = 8, SP = 64;
constexpr int D = L * SP;             // 512 feature dim
constexpr int TOK = BATCH * PCH * TT; // 256 tokens
constexpr int BPc = BATCH * PCH;      // 8
constexpr int NE = 100, NP = 112;     // NE padded to 7*16
constexpr int JTOT = D * D;           // 262144 contraction length
constexpr int JCH = 2048;             // split-K chunk
constexpr int NSPLIT = JTOT / JCH;    // 128
// LDS strides (padded vs 64 banks; even where b64 loads are used)
constexpr int KS_LD = 33, VS_LD = 66, A_LD = 114;

// ---------------- workspace layout (floats) ----------------
constexpr size_t OFF_W   = 0;                       // 3 * 512*512
constexpr size_t OFF_K   = OFF_W + 3 * (size_t)D * D;
constexpr size_t OFF_Q   = OFF_K + (size_t)TOK * D;
constexpr size_t OFF_V   = OFF_Q + (size_t)TOK * D;
constexpr size_t OFF_S   = OFF_V + (size_t)TOK * D; // 256*112
constexpr size_t OFF_U   = OFF_S + (size_t)TOK * NP; // 8*112
constexpr size_t OFF_H   = OFF_U + (size_t)BPc * NP; // 8*512*512

// =====================================================================
// TDM helper: 2D tensor tile (4-byte elements) -> LDS, optional LDS
// padding every 2^(pad_interval+1) dwords of pad_amount_code+1 dwords.
// D# packing per CDNA5 ISA ch.8 (groups 0/1; zero groups 2/3 for 2D).
// =====================================================================
__device__ __forceinline__ void tdm_load_2d(unsigned lds_addr, const void* gptr,
                                            unsigned tensor_d0, unsigned tensor_d1,
                                            unsigned tile_d0, unsigned tile_d1,
                                            unsigned row_stride_elems,
                                            unsigned pad_en, unsigned pad_interval,
                                            unsigned pad_amount_code) {
  unsigned long long ga = (unsigned long long)(size_t)gptr;
  u32x4 g0;
  g0[0] = 1u;                                              // count=1 (valid user D#)
  g0[1] = lds_addr;                                        // LDS byte address
  g0[2] = (unsigned)(ga & 0xFFFFFFFFu);                    // global_addr[31:0]
  g0[3] = (unsigned)((ga >> 32) & 0x01FFFFFFu) | (2u << 30); // addr[56:32] | type=2
  i32x8 g1;
  g1[0] = (int)((2u << 16) | (pad_en << 20) |
                (pad_interval << 22) | (pad_amount_code << 25)); // data_size=4B
  g1[1] = (int)((tensor_d0 & 0xFFFFu) << 16);              // tensor_dim0[15:0]
  g1[2] = (int)(((tensor_d0 >> 16) & 0xFFFFu) | ((tensor_d1 & 0xFFFFu) << 16));
  g1[3] = (int)(((tensor_d1 >> 16) & 0xFFFFu) | ((tile_d0 & 0xFFFFu) << 16));
  g1[4] = (int)(tile_d1 & 0xFFFFu);                        // tile_dim1 | tile_dim2=0
  g1[5] = (int)row_stride_elems;                           // tensor_dim0_stride[31:0]
  g1[6] = 0;                                               // stride[47:32] | dim1_stride lo
  g1[7] = 0;
  i32x4 z4 = {0, 0, 0, 0};
  i32x8 z8 = {0, 0, 0, 0, 0, 0, 0, 0};
  __builtin_amdgcn_tensor_load_to_lds(g0, g1, z4, z4, z8, 0);
}

__device__ __forceinline__ unsigned lds_offset(const void* p) {
  // LDS aperture maps flat addr[31:0] to the LDS byte offset (ISA 10.2)
  return (unsigned)(size_t)p;
}

// =====================================================================
// 1) Repack K/Q/V:  W[(c,x),(C,X)] = Wsrc[X,C,x,c]
// =====================================================================
__global__ __launch_bounds__(256) void repack_w(const float* __restrict__ Ksrc,
                                                const float* __restrict__ Qsrc,
                                                const float* __restrict__ Vsrc,
                                                float* __restrict__ Wdst) {
  int id = blockIdx.x * 256 + threadIdx.x;          // 3 * 262144 threads
  int mat = id >> 18;                               // /262144
  int e = id & (JTOT - 1);
  const float* src = (mat == 0) ? Ksrc : (mat == 1) ? Qsrc : Vsrc;
  int c = e & 7, x = (e >> 3) & 63, C = (e >> 9) & 7, X = (e >> 12) & 63;
  int fin = c * 64 + x, fout = C * 64 + X;
  Wdst[(size_t)mat * JTOT + fin * D + fout] = src[e];
}

// =====================================================================
// 2) k/q/v = x[256x512] @ W[512x512]   (WMMA fp32)
//    grid = (16 Mtiles, 8, 3), block = 128 (4 waves, 1 n-tile each)
// =====================================================================
__global__ __launch_bounds__(128) void kqv_gemm(const float* __restrict__ x,
                                                const float* __restrict__ W,
                                                float* __restrict__ ws) {
  int tid = threadIdx.x;
  int ml = tid & 15, half = (tid >> 4) & 1, wv = tid >> 5;
  int mt = blockIdx.x;                         // M tile (16 rows)
  int nt = blockIdx.y * 4 + wv;                // N tile
  int mat = blockIdx.z;
  const float* Wm = W + (size_t)mat * JTOT;
  float* out = ws + ((mat == 0) ? OFF_K : (mat == 1) ? OFF_Q : OFF_V);

  v8f acc = {};
  int row = mt * 16 + ml;
  int col = nt * 16 + ml;
  for (int kk = 0; kk < D; kk += 4) {
    v2f a = *(const v2f*)&x[(size_t)row * D + kk + 2 * half];
    v2f b;
    b.x = Wm[(size_t)(kk + 2 * half) * D + col];
    b.y = Wm[(size_t)(kk + 2 * half + 1) * D + col];
    acc = WMMA4(a, b, acc);
  }
#pragma unroll
  for (int i = 0; i < 8; ++i)
    out[(size_t)(mt * 16 + i + 8 * half) * D + col] = acc[i];
}

// =====================================================================
// 3) S[256,100] += O[256, Jchunk] @ AnoFlat[Jchunk, 16]   (split-K atomics)
//    O generated on the fly: O[t,j] = k[t, c*64+x] * v[t, kk*64+y]
//    grid = (128 ksplits, 7 ntiles), block = 256 (8 waves, 2 Mtiles each)
//    LDS: BsP[2048*16 pair-interleaved, manual] + kS/vS via TDM with
//    hardware LDS padding (strides 33 / 66 dwords).
// =====================================================================
__global__ __launch_bounds__(256) void bilinear_gemm(const float* __restrict__ Ano,
                                                     float* __restrict__ ws) {
  extern __shared__ float sm[];
  float* BsP = sm;                     // 2048*16, layout [(j>>1)][n][j&1]
  float* kS = BsP + JCH * 16;          // 256 rows * 33 (32 + 1 pad)
  float* vS = kS + TOK * KS_LD;        // 256 rows * 66 (64 + 2 pad)

  const float* kbuf = ws + OFF_K;
  const float* vbuf = ws + OFF_V;
  float* S = ws + OFF_S;

  int tid = threadIdx.x;
  int j0 = blockIdx.x * JCH;
  int n0 = blockIdx.y * 16;
  int c = j0 >> 15, kk_idx = (j0 >> 12) & 7, x0 = (j0 >> 6) & 63;

  // Tensor Data Mover: k slice (32x256, pad 32->33) and v slice
  // (64x256, pad 64->66). One wave issues; TENSORcnt tracks completion.
  if (tid < 32) {
    tdm_load_2d(lds_offset(kS), kbuf + c * 64 + x0,
                /*tensor_d0=*/32, /*tensor_d1=*/TOK,
                /*tile_d0=*/32, /*tile_d1=*/TOK,
                /*stride=*/D, /*pad_en=*/1, /*interval(32dw)=*/4, /*amount(1dw)=*/0);
    tdm_load_2d(lds_offset(vS), vbuf + kk_idx * 64,
                /*tensor_d0=*/64, /*tensor_d1=*/TOK,
                /*tile_d0=*/64, /*tile_d1=*/TOK,
                /*stride=*/D, /*pad_en=*/1, /*interval(64dw)=*/5, /*amount(2dw)=*/1);
  }

  // stage B tile, pair-interleaved so a lane's (K, K+1) pair is contiguous
  for (int r = tid; r < JCH; r += 256) {
    const float* src = Ano + (size_t)(j0 + r) * NE + n0;
    float* dst = BsP + (r >> 1) * 32 + (r & 1);
#pragma unroll
    for (int nn = 0; nn < 16; ++nn)
      dst[nn * 2] = (n0 + nn < NE) ? src[nn] : 0.0f;
  }
  if (tid < 32) __builtin_amdgcn_s_wait_tensorcnt(0);
  __syncthreads();

  int ml = tid & 15, half = (tid >> 4) & 1, wv = tid >> 5;
  int mt0 = wv * 2, mt1 = wv * 2 + 1;
  int t0 = mt0 * 16 + ml, t1 = mt1 * 16 + ml;

  const float* v0p = &vS[t0 * VS_LD + 2 * half];
  const float* v1p = &vS[t1 * VS_LD + 2 * half];

  v8f acc0 = {}, acc1 = {};
  for (int xi = 0; xi < 32; ++xi) {
    float kv0 = kS[t0 * KS_LD + xi];
    float kv1 = kS[t1 * KS_LD + xi];
    int jbase2 = xi * 32 + half;       // (js>>1) + half at ys=0
#pragma unroll 4
    for (int ys = 0; ys < 64; ys += 4) {
      v2f b = *(const v2f*)&BsP[(jbase2 + (ys >> 1)) * 32 + ml * 2];
      v2f vv0 = *(const v2f*)&v0p[ys];
      v2f vv1 = *(const v2f*)&v1p[ys];
      v2f a0, a1;
      a0.x = kv0 * vv0.x; a0.y = kv0 * vv0.y;
      a1.x = kv1 * vv1.x; a1.y = kv1 * vv1.y;
      acc0 = WMMA4(a0, b, acc0);
      acc1 = WMMA4(a1, b, acc1);
    }
  }
#pragma unroll
  for (int i = 0; i < 8; ++i) {
    atomicAdd(&S[(size_t)(mt0 * 16 + i + 8 * half) * NP + n0 + ml], acc0[i]);
    atomicAdd(&S[(size_t)(mt1 * 16 + i + 8 * half) * NP + n0 + ml], acc1[i]);
  }
}

// =====================================================================
// 4) scan: u = S[:,30] + sum_j Ann^(2^(j-2)) @ S[:, 29-j...]
//    single workgroup, 512 threads (16 waves). A@A via WMMA on LDS
//    ping-pong buffers (112x114, even stride -> b64 A-fragment loads).
// =====================================================================
__global__ __launch_bounds__(512) void scan_kernel(const float* __restrict__ Ann,
                                                   float* __restrict__ ws) {
  extern __shared__ float sm[];
  float* A0 = sm;                 // 112*114
  float* A1 = A0 + NP * A_LD;     // 112*114
  const float* S = ws + OFF_S;
  float* uBuf = ws + OFF_U;

  int tid = threadIdx.x;
  // init padded Ann
  for (int idx = tid; idx < NP * A_LD; idx += 512) {
    int r = idx / A_LD, cc = idx % A_LD;
    A0[idx] = (r < NE && cc < NE) ? Ann[r * NE + cc] : 0.0f;
  }
  // u slots: s0 = tid (if < 896), s1 = tid + 512 (if < 896)
  int s0 = tid, s1 = tid + 512;
  float u0 = 0.0f, u1 = 0.0f;
  if (s0 < BPc * NP) u0 = S[(size_t)((s0 / NP) * TT + (TT - 2)) * NP + (s0 % NP)];
  if (s1 < BPc * NP) u1 = S[(size_t)((s1 / NP) * TT + (TT - 2)) * NP + (s1 % NP)];
  __syncthreads();

  int ml = tid & 15, half = (tid >> 4) & 1, wv = tid >> 5;
  float* Ac = A0;
  float* An = A1;
  for (int iter = 0; iter < TT - 2; ++iter) {
    int t = (TT - 3) - iter;
    // u += Ac @ S[:, t]
    if (s0 < BPc * NP) {
      int bp = s0 / NP, n = s0 % NP;
      const float* Sr = S + (size_t)(bp * TT + t) * NP;
      const float* Ar = Ac + n * A_LD;
      float acc = 0.0f;
      for (int m = 0; m < NP; ++m) acc += Ar[m] * Sr[m];
      u0 += acc;
    }
    if (s1 < BPc * NP) {
      int bp = s1 / NP, n = s1 % NP;
      const float* Sr = S + (size_t)(bp * TT + t) * NP;
      const float* Ar = Ac + n * A_LD;
      float acc = 0.0f;
      for (int m = 0; m < NP; ++m) acc += Ar[m] * Sr[m];
      u1 += acc;
    }
    // An = Ac @ Ac  (7x7 tiles of 16x16, WMMA), skip on last iteration
    if (iter < TT - 3) {
      for (int tile = wv; tile < 49; tile += 16) {
        int i0 = (tile / 7) * 16, jj0 = (tile % 7) * 16;
        v8f acc = {};
        for (int kk = 0; kk < NP; kk += 4) {
          v2f a = *(const v2f*)&Ac[(i0 + ml) * A_LD + kk + 2 * half];
          v2f b;
          b.x = Ac[(kk + 2 * half) * A_LD + jj0 + ml];
          b.y = Ac[(kk + 2 * half + 1) * A_LD + jj0 + ml];
          acc = WMMA4(a, b, acc);
        }
#pragma unroll
        for (int i = 0; i < 8; ++i)
          An[(i0 + i + 8 * half) * A_LD + jj0 + ml] = acc[i];
      }
    }
    __syncthreads();
    float* tmp = Ac; Ac = An; An = tmp;
  }
  if (s0 < BPc * NP) uBuf[s0] = u0;
  if (s1 < BPc * NP) uBuf[s1] = u1;
}

// =====================================================================
// 5) H[bp, (c,x), (k,y)] = Aoo[c,k,x,y]*k_last[bp,c,x]*v_last[bp,k,y]
//                          + sum_n Aon[c,k,x,y,n] * u[bp,n]
//    one thread per (c,k,x,y); Aon n-vector is contiguous.
// =====================================================================
__global__ __launch_bounds__(256) void build_h(const float* __restrict__ Aoo,
                                               const float* __restrict__ Aon,
                                               float* __restrict__ ws) {
  __shared__ float uS[BPc * NE];
  const float* uBuf = ws + OFF_U;
  const float* kbuf = ws + OFF_K;
  const float* vbuf = ws + OFF_V;
  float* H = ws + OFF_H;

  int tid = threadIdx.x;
#pragma unroll
  for (int o = 0; o < BPc * NE; o += 256) {
    int s = o + tid;
    if (s < BPc * NE) uS[s] = uBuf[(s / NE) * NP + (s % NE)];
  }
  __syncthreads();

  int e = blockIdx.x * 256 + tid;            // (c,k,x,y) flat, 262144 total
  int c = (e >> 15) & 7, kk_idx = (e >> 12) & 7, x = (e >> 6) & 63, y = e & 63;
  int r = c * 64 + x, col = kk_idx * 64 + y;

  float aoo = Aoo[e];
  const float* an = Aon + (size_t)e * NE;
  float acc[BPc];
#pragma unroll
  for (int bp = 0; bp < BPc; ++bp) {
    float kl = kbuf[(size_t)(bp * TT + TT - 1) * D + r];
    float vl = vbuf[(size_t)(bp * TT + TT - 1) * D + col];
    acc[bp] = aoo * kl * vl;
  }
  for (int n = 0; n < NE; ++n) {
    float av = an[n];
#pragma unroll
    for (int bp = 0; bp < BPc; ++bp) acc[bp] += av * uS[bp * NE + n];
  }
#pragma unroll
  for (int bp = 0; bp < BPc; ++bp)
    H[(size_t)bp * JTOT + (size_t)r * D + col] = acc[bp];
}

// =====================================================================
// 6) y[bp, t, col] = q[bp*32+t, :] @ H[bp][:, col]   (WMMA fp32)
//    grid = (32 col-tiles, 8 bp), block = 64 (2 waves, 1 Mtile each)
// =====================================================================
__global__ __launch_bounds__(64) void final_gemm(const float* __restrict__ ws,
                                                 float* __restrict__ out) {
  const float* qbuf = ws + OFF_Q;
  const float* H = ws + OFF_H;
  int tid = threadIdx.x;
  int ml = tid & 15, half = (tid >> 4) & 1, mt = tid >> 5;
  int bp = blockIdx.y;
  int col = blockIdx.x * 16 + ml;
  const float* Hb = H + (size_t)bp * JTOT;
  int row = bp * TT + mt * 16 + ml;

  v8f acc = {};
  for (int kk = 0; kk < D; kk += 4) {
    v2f a = *(const v2f*)&qbuf[(size_t)row * D + kk + 2 * half];
    v2f b;
    b.x = Hb[(size_t)(kk + 2 * half) * D + col];
    b.y = Hb[(size_t)(kk + 2 * half + 1) * D + col];
    acc = WMMA4(a, b, acc);
  }
#pragma unroll
  for (int i = 0; i < 8; ++i)
    out[(size_t)(bp * TT + mt * 16 + i + 8 * half) * D + col] = acc[i];
}

// =====================================================================
extern "C" void kernel_launch(void* const* d_in, const int* in_sizes, int n_in,
                              void* d_out, int out_size, void* d_ws, size_t ws_size,
                              hipStream_t stream) {
  const float* x   = (const float*)d_in[0];
  const float* Kw  = (const float*)d_in[1];
  const float* Qw  = (const float*)d_in[2];
  const float* Vw  = (const float*)d_in[3];
  const float* Aoo = (const float*)d_in[4];
  const float* Ann = (const float*)d_in[5];
  const float* Aon = (const float*)d_in[6];
  const float* Ano = (const float*)d_in[7];
  float* ws = (float*)d_ws;
  float* out = (float*)d_out;

  // 1) repack weight matrices
  repack_w<<<3 * JTOT / 256, 256, 0, stream>>>(Kw, Qw, Vw, ws + OFF_W);
  // 2) k, q, v projections
  kqv_gemm<<<dim3(TOK / 16, D / 64, 3), 128, 0, stream>>>(x, ws + OFF_W, ws);
  // 3) zero S, then split-K bilinear GEMM with atomics (TDM-staged k/v)
  (void)hipMemsetAsync(ws + OFF_S, 0, (size_t)TOK * NP * sizeof(float), stream);
  size_t lds3 = (size_t)(JCH * 16 + TOK * KS_LD + TOK * VS_LD) * sizeof(float);
  bilinear_gemm<<<dim3(NSPLIT, (NP / 16)), 256, lds3, stream>>>(Ano, ws);
  // 4) sequential scan with repeated squaring (single WG)
  size_t lds4 = (size_t)(2 * NP * A_LD) * sizeof(float);
  scan_kernel<<<1, 512, lds4, stream>>>(Ann, ws);
  // 5) build H
  build_h<<<JTOT / 256, 256, 0, stream>>>(Aoo, Aon, ws);
  // 6) final projection into d_out
  final_gemm<<<dim3(D / 16, BPc), 64, 0, stream>>>(ws, out);
}